// HIME_927712936544
// MI455X (gfx1250) — compile-verified
//
#include <hip/hip_runtime.h>
#include <hip/hip_bf16.h>

// ---------------------------------------------------------------------------
// HIME node/tag distance kernel for MI455X (gfx1250, wave32, WMMA).
//
// dist[b,e] = -<node_tables[e, node[b], :], tag_table[tag[b], :]>  (D=32)
// dmin/argmin over e=0..3; pos: activate[node,argmin]+=1, loss+=softplus(dmin)
//                          neg: loss += softplus(-dmin)
//
// Dot products computed with V_WMMA_F32_16X16X4_F32: a 16-element batch tile
// forms A(16x4)=tag chunks, B(4x16)=node chunks; the 16x16 D diagonal is the
// per-element dot. Full f32 precision, 8 chained WMMAs per expert per tile.
// ---------------------------------------------------------------------------

typedef float v2f __attribute__((ext_vector_type(2)));
typedef float v8f __attribute__((ext_vector_type(8)));

#define HIME_NODE 100000
#define HIME_D    32
#define HIME_E    4

__global__ void HIME_init_out(const float* __restrict__ activate,
                              float* __restrict__ out, int n_act) {
    int i = blockIdx.x * blockDim.x + threadIdx.x;
    if (i == 0) out[0] = 0.0f;                // loss accumulator
    if (i < n_act) out[1 + i] = activate[i];  // activate_new starts as activate
}

__device__ __forceinline__ float hime_pick8(v8f v, int r) {
    // select component r (r uniform-per-lane but not compile-time): cndmask chain
    float x = v[0];
    x = (r == 1) ? v[1] : x;
    x = (r == 2) ? v[2] : x;
    x = (r == 3) ? v[3] : x;
    x = (r == 4) ? v[4] : x;
    x = (r == 5) ? v[5] : x;
    x = (r == 6) ? v[6] : x;
    x = (r == 7) ? v[7] : x;
    return x;
}

__global__ __launch_bounds__(256) void HIME_dist_kernel(
    const float* __restrict__ tag_table,    // (TAG, 32)
    const float* __restrict__ node_tables,  // (4, NODE, 32)
    const int*   __restrict__ node_list,    // (B,)
    const int*   __restrict__ tag_list,     // (B,)
    float* __restrict__ out_loss,           // d_out[0]
    float* __restrict__ out_activate,       // d_out+1, (NODE, 4)
    int B, int is_pos)
{
    const int lane   = threadIdx.x & 31;
    const int wid    = (blockIdx.x * blockDim.x + threadIdx.x) >> 5;
    const int nwaves = (gridDim.x * blockDim.x) >> 5;
    const int ntiles = (B + 15) >> 4;

    const int m = lane & 15;   // batch row within tile this lane loads
    const int h = lane >> 4;   // K-half (ISA A/B f32 layout: lanes16-31 hold K=2,3)

    // C/D 16x16 f32 layout: diag element m is held by lane m (vgpr m) for m<8,
    // and lane 16+m (vgpr m-8) for m>=8  ->  owner lanes are 0-7 and 24-31,
    // and each owner lane's diag is its own batch element (m == lane&15).
    const bool own = (lane < 8) || (lane >= 24);
    const int  r   = (lane < 8) ? lane : (lane - 24);

    float loss = 0.0f;

    for (int t = wid; t < ntiles; t += nwaves) {
        int  b     = t * 16 + m;
        bool valid = (b < B);
        int  bc    = valid ? b : (B - 1);

        int nid = node_list[bc];
        int tid = tag_list[bc];

        // A chunks: tag_emb[b_m, 4c + 2h + {0,1}]  (8 x float2 per lane)
        const float* tr = tag_table + (size_t)tid * HIME_D;
        v2f a[8];
        #pragma unroll
        for (int c = 0; c < 8; ++c)
            a[c] = *(const v2f*)(tr + 4 * c + 2 * h);

        float dist[HIME_E];
        #pragma unroll
        for (int e = 0; e < HIME_E; ++e) {
            const float* nr =
                node_tables + ((size_t)e * HIME_NODE + (size_t)nid) * HIME_D;
            v8f acc = {0.f, 0.f, 0.f, 0.f, 0.f, 0.f, 0.f, 0.f};
            #pragma unroll
            for (int c = 0; c < 8; ++c) {
                v2f bb = *(const v2f*)(nr + 4 * c + 2 * h);
                // D(16x16,f32) += A(16x4,f32) x B(4x16,f32)
                acc = __builtin_amdgcn_wmma_f32_16x16x4_f32(
                    /*neg_a=*/false, a[c], /*neg_b=*/false, bb,
                    /*c_mod=*/(short)0, acc,
                    /*reuse_a=*/false, /*reuse_b=*/false);
            }
            dist[e] = -hime_pick8(acc, r);  // dist = -dot (diag element)
        }

        if (own && valid) {
            // first-occurrence argmin over e (matches jnp.argmin tie rule)
            float dmin = dist[0];
            int   imin = 0;
            #pragma unroll
            for (int e = 1; e < HIME_E; ++e)
                if (dist[e] < dmin) { dmin = dist[e]; imin = e; }

            // pos: softplus(dmin) ; neg: softplus(-dmin)   (stable form)
            float x  = is_pos ? dmin : -dmin;
            float ax = fabsf(x);
            loss += fmaxf(x, 0.0f) + __logf(1.0f + __expf(-ax));

            if (is_pos)
                atomicAdd(&out_activate[(size_t)nid * HIME_E + imin], 1.0f);
        }
    }

    // wave32 reduction of loss, one global f32 atomic per wave
    #pragma unroll
    for (int off = 16; off >= 1; off >>= 1)
        loss += __shfl_xor(loss, off, 32);
    if (lane == 0 && loss != 0.0f)
        atomicAdd(out_loss, loss);
    else if (lane == 0)
        atomicAdd(out_loss, 0.0f);  // keep deterministic op count (cheap)
}

extern "C" void kernel_launch(void* const* d_in, const int* in_sizes, int n_in,
                              void* d_out, int out_size, void* d_ws, size_t ws_size,
                              hipStream_t stream) {
    const float* tag_table   = (const float*)d_in[0];
    const float* node_tables = (const float*)d_in[1];
    const float* activate    = (const float*)d_in[2];
    const int*   pos_node    = (const int*)d_in[3];
    const int*   pos_tag     = (const int*)d_in[4];
    const int*   neg_node    = (const int*)d_in[5];
    const int*   neg_tag     = (const int*)d_in[6];

    float* out   = (float*)d_out;
    int    n_act = in_sizes[2];  // NODE*E = 400000
    int    B     = in_sizes[3];  // 524288

    HIME_init_out<<<(n_act + 256) / 256, 256, 0, stream>>>(activate, out, n_act);

    // L2-gather-bound: B/16 = 32768 tiles per side; 2048 blocks x 8 waves
    // -> 16384 waves -> 2 tiles/wave. Plenty of latency hiding.
    int blocks = 2048;
    HIME_dist_kernel<<<blocks, 256, 0, stream>>>(
        tag_table, node_tables, pos_node, pos_tag, out, out + 1, B, 1);
    HIME_dist_kernel<<<blocks, 256, 0, stream>>>(
        tag_table, node_tables, neg_node, neg_tag, out, out + 1, B, 0);
}